// RayMarcher_23648089932181
// MI455X (gfx1250) — compile-verified
//
#include <hip/hip_runtime.h>
#include <math.h>

#define NS 96           // samples per ray
#define NC 32           // channels
#define RPB 4           // rays per block (one wave32 per ray)
#define DELTA_INF 1e10f
#define EPSW 1e-10f

typedef float v2f __attribute__((ext_vector_type(2)));
typedef float v8f __attribute__((ext_vector_type(8)));

// ---------------- global depth min/max (for jnp.clip bounds) ----------------

__global__ void mm_init_kernel(unsigned int* ws) {
    ws[0] = 0xFFFFFFFFu;   // running min (uint-ordered floats, all depths > 0)
    ws[1] = 0u;            // running max
}

__global__ void mm_reduce_kernel(const float* __restrict__ depths,
                                 unsigned int* __restrict__ ws, int nrays) {
    int r = blockIdx.x * blockDim.x + threadIdx.x;
    if (r >= nrays) return;
    // depths sorted ascending along sample axis
    float dmin = depths[(size_t)r * NS];
    float dmax = depths[(size_t)r * NS + NS - 1];
    atomicMin(&ws[0], __float_as_uint(dmin));
    atomicMax(&ws[1], __float_as_uint(dmax));
}

// ---------------- main ray-march kernel ----------------

__device__ __forceinline__ float softplus_f(float x) {
    // numerically stable: max(x,0) + log1p(exp(-|x|))
    return fmaxf(x, 0.f) + log1pf(expf(-fabsf(x)));
}

__global__ __launch_bounds__(32 * RPB)
void raymarch_kernel(const float* __restrict__ colors,
                     const float* __restrict__ densities,
                     const float* __restrict__ depths,
                     float* __restrict__ rgb_out,
                     float* __restrict__ depth_out,
                     float* __restrict__ weights_out,
                     const unsigned int* __restrict__ ws) {
    __shared__ __align__(16) float lds_col[RPB][NS * NC];  // 48 KB
    __shared__ __align__(16) float lds_w[RPB][NS];         // 1.5 KB

    const int lane = threadIdx.x & 31;
    const int wid  = threadIdx.x >> 5;
    const size_t ray = (size_t)blockIdx.x * RPB + wid;

    const float* col = colors    + ray * (size_t)(NS * NC);
    const float* den = densities + ray * NS;
    const float* dep = depths    + ray * NS;
    float* colLds = &lds_col[wid][0];
    float* wLds   = &lds_w[wid][0];

    // ---- stage the 12KB color tile into LDS (async DMA path on CDNA5) ----
#if defined(__gfx1250__) && __has_builtin(__builtin_amdgcn_global_load_async_to_lds_b128)
    {
        typedef int v4i __attribute__((vector_size(16)));
        typedef __attribute__((address_space(1))) v4i gv4i;   // prints as "__device__"
        typedef __attribute__((address_space(3))) v4i lv4i;   // prints as "__shared__"
        gv4i* g = (gv4i*)col;        // cast away const; builtin param is non-const
        lv4i* l = (lv4i*)colLds;
#pragma unroll
        for (int i = 0; i < (NS * NC * 4) / (32 * 16); ++i) {   // 24 chunks of 512B
            int idx = i * 32 + lane;                            // 16B elements
            __builtin_amdgcn_global_load_async_to_lds_b128(g + idx, l + idx, 0, 0);
        }
    }
#else
    {
        const float4* gsrc = (const float4*)col;
        float4* ldst = (float4*)colLds;
#pragma unroll
        for (int i = 0; i < (NS * NC) / (4 * 32); ++i)
            ldst[i * 32 + lane] = gsrc[i * 32 + lane];
    }
#endif

    // ---- phase 1: weights via wave-parallel exclusive cumprod scan ----
    // lane handles samples s0, s0+1, s0+2 (contiguous trio)
    const int s0 = 3 * lane;
    float d0 = dep[s0], d1 = dep[s0 + 1], d2 = dep[s0 + 2];
    float d3 = (s0 + 3 < NS) ? dep[s0 + 3] : 0.f;

    float dlA = d1 - d0;
    float dlB = d2 - d1;
    float dlC = (lane == 31) ? DELTA_INF : (d3 - d2);

    float sg0 = softplus_f(den[s0]     - 1.f);   // VOLUME_INIT = -1
    float sg1 = softplus_f(den[s0 + 1] - 1.f);
    float sg2 = softplus_f(den[s0 + 2] - 1.f);

    float al0 = 1.f - expf(-dlA * sg0);
    float al1 = 1.f - expf(-dlB * sg1);
    float al2 = 1.f - expf(-dlC * sg2);

    float q0 = 1.f - al0 + EPSW;
    float q1 = 1.f - al1 + EPSW;
    float q2 = 1.f - al2 + EPSW;

    // exclusive product scan across lanes of p = q0*q1*q2
    float scan = q0 * q1 * q2;
#pragma unroll
    for (int off = 1; off < 32; off <<= 1) {
        float t = __shfl_up(scan, off, 32);
        if (lane >= off) scan *= t;
    }
    float excl = __shfl_up(scan, 1, 32);
    float T = (lane == 0) ? 1.f : excl;

    float w0 = al0 * T;
    float w1 = al1 * (T * q0);
    float w2 = al2 * (T * q0 * q1);

    // weights output (global) + LDS copy for the WMMA phase
    float* wout = weights_out + ray * NS + s0;
    wout[0] = w0; wout[1] = w1; wout[2] = w2;
    wLds[s0] = w0; wLds[s0 + 1] = w1; wLds[s0 + 2] = w2;

    // wsum and depth accumulation (tree reduction, deterministic)
    float wpart = w0 + w1 + w2;
    float dpart = w0 * d0 + w1 * d1 + w2 * d2;
#pragma unroll
    for (int off = 16; off; off >>= 1) {
        wpart += __shfl_xor(wpart, off, 32);
        dpart += __shfl_xor(dpart, off, 32);
    }
    if (lane == 0) {
        float df = dpart / wpart;                 // 0/0 -> NaN
        if (isnan(df)) df = INFINITY;             // where(isnan) -> inf
        float gmin = __uint_as_float(ws[0]);
        float gmax = __uint_as_float(ws[1]);
        df = fminf(fmaxf(df, gmin), gmax);        // clip (inf -> gmax)
        depth_out[ray] = df * 2.f - 1.f;
    }

    // ---- wait for async color tile ----
#if defined(__gfx1250__) && __has_builtin(__builtin_amdgcn_global_load_async_to_lds_b128)
#if __has_builtin(__builtin_amdgcn_s_wait_asynccnt)
    __builtin_amdgcn_s_wait_asynccnt(0);
#else
    asm volatile("s_wait_asynccnt 0" ::: "memory");
#endif
    asm volatile("" ::: "memory");
#endif

    // ---- phase 2: rgb[c] = sum_s w[s]*col[s,c] via chained f32 WMMA ----
    // A (16x4, w replicated over M): lanes 0-15 hold K={0,1}, lanes 16-31 K={2,3}
    // B (4x16):  lane n%16 holds cols; VGPR0=K(2h), VGPR1=K(2h+1)
    const int h = lane >> 4;        // half: 0 or 1
    const int n = lane & 15;        // N index within half
    v8f acc0 = {};                  // channels 0..15
    v8f acc1 = {};                  // channels 16..31
#pragma unroll
    for (int j = 0; j < NS / 4; ++j) {          // 24 K-chunks of 4 samples
        int sA = 4 * j + 2 * h;
        v2f a;  a.x  = wLds[sA];                 a.y  = wLds[sA + 1];
        v2f b0; b0.x = colLds[sA * NC + n];      b0.y = colLds[(sA + 1) * NC + n];
        v2f b1; b1.x = colLds[sA * NC + n + 16]; b1.y = colLds[(sA + 1) * NC + n + 16];
        acc0 = __builtin_amdgcn_wmma_f32_16x16x4_f32(false, a, false, b0,
                                                     (short)0, acc0, false, false);
        acc1 = __builtin_amdgcn_wmma_f32_16x16x4_f32(false, a, false, b1,
                                                     (short)0, acc1, false, false);
    }
    // all D rows equal (A rows replicated): any lane's VGPR0 = rgb[n (+16)]
    float rgb = (lane < 16) ? acc0[0] : acc1[0];   // channel == lane
    rgb_out[ray * NC + lane] = rgb * 2.f - 1.f;
}

// ---------------- launch ----------------

extern "C" void kernel_launch(void* const* d_in, const int* in_sizes, int n_in,
                              void* d_out, int out_size, void* d_ws, size_t ws_size,
                              hipStream_t stream) {
    const float* colors    = (const float*)d_in[0];
    const float* densities = (const float*)d_in[1];
    const float* depths    = (const float*)d_in[2];

    int nrays = in_sizes[1] / NS;   // densities is [B,R,S,1] -> B*R rays

    float* rgb_out     = (float*)d_out;                       // [B*R, 32]
    float* depth_out   = rgb_out + (size_t)nrays * NC;        // [B*R]
    float* weights_out = depth_out + nrays;                   // [B*R, 96]

    unsigned int* mm = (unsigned int*)d_ws;

    mm_init_kernel<<<1, 1, 0, stream>>>(mm);
    mm_reduce_kernel<<<(nrays + 255) / 256, 256, 0, stream>>>(depths, mm, nrays);
    raymarch_kernel<<<nrays / RPB, 32 * RPB, 0, stream>>>(
        colors, densities, depths, rgb_out, depth_out, weights_out, mm);
}